// Sinkhorn_68487548502073
// MI455X (gfx1250) — compile-verified
//
#include <hip/hip_runtime.h>
#include <stdint.h>

// Sinkhorn (B=8, M=N=2048, reg=1, 100 iters) for MI455X / gfx1250.
//
// Linear-domain reformulation: K = exp(-cost); beta0 = 1;
//   alpha = (1/M) ./ (K beta) ; beta = (1/N) ./ (K^T alpha)   (x100)
//   plan = K .* (alpha beta^T); cost_b = sum(plan .* cost)
// K (134 MB) is precomputed once into d_ws and stays L2-resident (192 MB L2),
// so the 200 matrix sweeps stream from L2, not HBM.
//
// Row update uses V_WMMA_F32_16X16X4_F32 (16-row strip per wave32), fed from
// LDS tiles staged by the Tensor Data Mover with hardware row padding
// (stride-65 dwords -> conflict-free ds reads). Column update is a coalesced
// VALU streaming kernel.

#define SB 8
#define SM 2048
#define SN 2048
#define SMN (SM * SN)
#define SITERS 100

typedef __attribute__((ext_vector_type(2))) float v2f;
typedef __attribute__((ext_vector_type(8))) float v8f;
typedef __attribute__((ext_vector_type(4))) unsigned int u32x4;
typedef __attribute__((ext_vector_type(4))) int i32x4;
typedef __attribute__((ext_vector_type(8))) int i32x8;

#if defined(__has_builtin)
#if __has_builtin(__builtin_amdgcn_tensor_load_to_lds) && \
    __has_builtin(__builtin_amdgcn_s_wait_tensorcnt)
#define SINK_HAS_TDM 1
#endif
#endif
#ifndef SINK_HAS_TDM
#define SINK_HAS_TDM 0
#endif

// Generic LDS pointer: addr[63:32] = SHARED_BASE aperture, addr[31:0] = LDS
// byte offset (CDNA5 ISA 10.2 aperture rules). Truncation yields the D# lds_addr.
static __device__ __forceinline__ unsigned sink_lds_off(void* p) {
  return (unsigned)(uintptr_t)p;
}

#if SINK_HAS_TDM
// TDM descriptor for a 16x64 fp32 tile of a 2048x2048 matrix, with LDS row
// padding: pad_interval=5 (after every 64 dwords) pad_amount=0 (1 dword)
// -> LDS row stride 65 dwords -> ds_load gathers are bank-conflict-free.
static __device__ __forceinline__ void tdm_load_tile16x64(const float* gsrc,
                                                          unsigned lds_byte) {
  unsigned long long ga = (unsigned long long)(uintptr_t)gsrc;
  u32x4 g0;
  g0[0] = 1u;                                   // count=1 (valid), user mode
  g0[1] = lds_byte;                             // lds_addr
  g0[2] = (unsigned)ga;                         // global_addr[31:0]
  g0[3] = (unsigned)((ga >> 32) & 0x1FFFFFFull) // global_addr[56:32]
          | (2u << 30);                         // type=2 ("image")
  i32x8 g1;
  g1[0] = (int)((2u << 16)     // data_size = 4 bytes
                | (1u << 20)   // pad_enable
                | (5u << 22)); // pad_interval = 64 dwords (pad_amount=0 -> 1 dw)
  g1[1] = (int)(((unsigned)SN & 0xFFFFu) << 16);               // tensor_dim0 lo
  g1[2] = (int)((((unsigned)SN >> 16) & 0xFFFFu) |
                (((unsigned)SM & 0xFFFFu) << 16));             // dim0 hi|dim1 lo
  g1[3] = (int)((((unsigned)SM >> 16) & 0xFFFFu) | (64u << 16)); // |tile_dim0=64
  g1[4] = (int)16u;                                            // tile_dim1=16
  g1[5] = (int)SN;                                             // dim0_stride lo
  g1[6] = 0;                                  // dim0_stride hi | dim1_stride lo
  g1[7] = (int)(((unsigned long long)SMN) >> 16);       // dim1_stride[47:16]
  i32x4 z4 = {0, 0, 0, 0};
#if __clang_major__ >= 23
  i32x8 z8 = {0, 0, 0, 0, 0, 0, 0, 0};
  __builtin_amdgcn_tensor_load_to_lds(g0, g1, z4, z4, z8, 0);
#else
  __builtin_amdgcn_tensor_load_to_lds(g0, g1, z4, z4, 0);
#endif
}
#endif

// ---------------------------------------------------------------------------
// K = exp(-cost); beta = 1
__global__ __launch_bounds__(256) void sink_exp_init(
    const float* __restrict__ cost, float* __restrict__ K,
    float* __restrict__ beta) {
  size_t idx = (size_t)blockIdx.x * blockDim.x + threadIdx.x;
  const size_t stride = (size_t)gridDim.x * blockDim.x;
  const size_t total = (size_t)SB * SMN;
  for (size_t i = idx; i < total; i += stride) K[i] = __expf(-cost[i]);
  if (idx < (size_t)SB * SN) beta[idx] = 1.0f;
}

// ---------------------------------------------------------------------------
// alpha[b,i] = (1/M) / sum_j K[b,i,j] * beta[b,j]
// One wave32 per 16-row strip. TDM double-buffers 16x64 tiles into padded LDS;
// reduction runs on V_WMMA_F32_16X16X4_F32 with beta replicated across B's
// 16 columns (each D column holds the strip matvec).
__global__ __launch_bounds__(32) void sink_row_update(
    const float* __restrict__ K, const float* __restrict__ beta,
    float* __restrict__ alpha) {
  constexpr int STRIPS = SM / 16;
  const int b = blockIdx.x / STRIPS;
  const int r0 = (blockIdx.x % STRIPS) * 16;
  const float* Kb = K + (size_t)b * SMN + (size_t)r0 * SN;
  const float* bb = beta + b * SN;
  const int lane = threadIdx.x;
  const int m = lane & 15;          // A-matrix row for this lane
  const int kh = (lane >> 4) << 1;  // K offset: lanes 0-15 -> k0/k1, 16-31 -> k2/k3

  __shared__ float tile[2][16 * 65];  // 65-dword row stride (TDM pad)

  v8f acc = {0.f, 0.f, 0.f, 0.f, 0.f, 0.f, 0.f, 0.f};
  constexpr int TILES = SN / 64;

#if SINK_HAS_TDM
  tdm_load_tile16x64(Kb, sink_lds_off(&tile[0][0]));
#endif
  for (int t = 0; t < TILES; ++t) {
    const int buf = t & 1;
#if SINK_HAS_TDM
    if (t + 1 < TILES) {
      tdm_load_tile16x64(Kb + (t + 1) * 64, sink_lds_off(&tile[buf ^ 1][0]));
      __builtin_amdgcn_s_wait_tensorcnt(1);  // tile t complete (in-order TDM)
    } else {
      __builtin_amdgcn_s_wait_tensorcnt(0);
    }
#else
    {  // cooperative synchronous staging with the same padded layout
      const float* g = Kb + t * 64;
#pragma unroll
      for (int r = 0; r < 16; ++r) {
        tile[buf][r * 65 + lane] = g[(size_t)r * SN + lane];
        tile[buf][r * 65 + lane + 32] = g[(size_t)r * SN + lane + 32];
      }
    }
#endif
    const float* lt = &tile[buf][0];
    const float* bc = bb + t * 64;
#pragma unroll
    for (int kk = 0; kk < 16; ++kk) {
      const int kb = kk * 4 + kh;
      v2f a;  // 16x4 f32 A operand: VGPR0 = k0/k2, VGPR1 = k1/k3 per half-wave
      a.x = lt[m * 65 + kb];
      a.y = lt[m * 65 + kb + 1];
      v2f v;  // 4x16 f32 B operand: beta[k] replicated across all 16 columns
      v.x = bc[kb];
      v.y = bc[kb + 1];
      acc = __builtin_amdgcn_wmma_f32_16x16x4_f32(false, a, false, v, (short)0,
                                                  acc, false, false);
    }
  }
  // D layout: VGPR r holds M=r (lanes 0-15) and M=r+8 (lanes 16-31); every
  // column is identical, so lane 0 / lane 16 hold the full strip result.
  const float la = 1.0f / (float)SM;
  if (lane == 0) {
#pragma unroll
    for (int r = 0; r < 8; ++r) alpha[b * SM + r0 + r] = la / acc[r];
  } else if (lane == 16) {
#pragma unroll
    for (int r = 0; r < 8; ++r) alpha[b * SM + r0 + 8 + r] = la / acc[r];
  }
}

// ---------------------------------------------------------------------------
// beta[b,j] = (1/N) / sum_i K[b,i,j] * alpha[b,i]
// Block = 256 threads: 32 columns x 8 row-groups; coalesced 128B row segments,
// alpha[i] is wave-uniform (scalar load), LDS combine of the 8 partials.
__global__ __launch_bounds__(256) void sink_col_update(
    const float* __restrict__ K, const float* __restrict__ alpha,
    float* __restrict__ beta) {
  const int b = blockIdx.y;
  const int tx = threadIdx.x & 31;
  const int ty = threadIdx.x >> 5;  // 0..7
  const int j = blockIdx.x * 32 + tx;
  const float* Kb = K + (size_t)b * SMN;
  const float* al = alpha + b * SM;
  constexpr int ROWS = SM / 8;  // 256 rows per group
  const int i0 = ty * ROWS;
  float acc = 0.0f;
  for (int i = i0; i < i0 + ROWS; ++i) {
    if ((i & 31) == 0)
      __builtin_prefetch(&Kb[(size_t)(i + 32) * SN + j], 0, 0);
    acc = __builtin_fmaf(Kb[(size_t)i * SN + j], al[i], acc);
  }
  __shared__ float red[256];
  red[threadIdx.x] = acc;
  __syncthreads();
  if (ty == 0) {
    float s = red[tx];
#pragma unroll
    for (int g = 1; g < 8; ++g) s += red[g * 32 + tx];
    beta[b * SN + j] = (1.0f / (float)SN) / s;
  }
}

// ---------------------------------------------------------------------------
__global__ void sink_zero_cost(float* __restrict__ costout) {
  if (threadIdx.x < SB) costout[threadIdx.x] = 0.0f;
}

// plan = K * alpha_i * beta_j (written to d_out) ; cost_b += plan * cost
__global__ __launch_bounds__(256) void sink_finalize(
    const float* __restrict__ K, const float* __restrict__ cost,
    const float* __restrict__ alpha, const float* __restrict__ beta,
    float* __restrict__ plan, float* __restrict__ costout) {
  const int b = blockIdx.y;
  const size_t base = (size_t)b * SMN;
  float lsum = 0.0f;
  for (size_t off = (size_t)blockIdx.x * blockDim.x + threadIdx.x;
       off < (size_t)SMN; off += (size_t)gridDim.x * blockDim.x) {
    const int i = (int)(off >> 11);    // / 2048
    const int j = (int)(off & 2047);   // % 2048
    const float p = K[base + off] * alpha[b * SM + i] * beta[b * SN + j];
    plan[base + off] = p;
    lsum = __builtin_fmaf(p, cost[base + off], lsum);
  }
  __shared__ float red[256];
  red[threadIdx.x] = lsum;
  __syncthreads();
  for (int s = 128; s > 0; s >>= 1) {
    if (threadIdx.x < s) red[threadIdx.x] += red[threadIdx.x + s];
    __syncthreads();
  }
  if (threadIdx.x == 0) atomicAdd(&costout[b], red[0]);
}

// ---------------------------------------------------------------------------
extern "C" void kernel_launch(void* const* d_in, const int* in_sizes, int n_in,
                              void* d_out, int out_size, void* d_ws,
                              size_t ws_size, hipStream_t stream) {
  (void)in_sizes; (void)n_in; (void)out_size; (void)ws_size;
  const float* cost = (const float*)d_in[0];

  // Workspace: K (134.2 MB, L2-resident), alpha (8x2048), beta (8x2048)
  float* K = (float*)d_ws;
  float* alpha = K + (size_t)SB * SMN;
  float* beta = alpha + (size_t)SB * SM;

  float* plan = (float*)d_out;                 // 8*2048*2048 fp32
  float* costout = plan + (size_t)SB * SMN;    // 8 fp32

  sink_exp_init<<<8192, 256, 0, stream>>>(cost, K, beta);

  for (int it = 0; it < SITERS; ++it) {
    sink_row_update<<<SB * (SM / 16), 32, 0, stream>>>(K, beta, alpha);
    sink_col_update<<<dim3(SN / 32, SB), 256, 0, stream>>>(K, alpha, beta);
  }

  sink_zero_cost<<<1, 32, 0, stream>>>(costout);
  sink_finalize<<<dim3(1024, SB), 256, 0, stream>>>(K, cost, alpha, beta, plan,
                                                    costout);
}